// Net_74174085202653
// MI455X (gfx1250) — compile-verified
//
#include <hip/hip_runtime.h>
#include <math.h>

// ---------------------------------------------------------------------------
// GCN (2-layer) for MI455X / gfx1250, wave32.
// Factored norm:  out[d] = dinv[d] * ( sum_{e: dst=d} y[src_e] + y[d] ),
// where y[i] = dinv[i] * (x @ W)[i].  Edge phase is a pure gather +
// global_atomic_add_f32 scatter (L2-resident: per-node tables are 16-32 MB,
// well inside the 192 MB L2); dense GEMMs use f32 WMMA (16x16x4), with the
// dinv row-scaling folded into the A operand so the epilogue is pure
// coalesced stores.
// ---------------------------------------------------------------------------

typedef __attribute__((ext_vector_type(2))) float v2f;
typedef __attribute__((ext_vector_type(8))) float v8f;

#define TPB 256

// ---------------- utility: fill a float region -----------------------------
__global__ void k_fill_f32(float* __restrict__ p, float v, int n) {
    int i = blockIdx.x * blockDim.x + threadIdx.x;
    if (i < n) p[i] = v;
}

// ---------------- degree: deg[dst] += 1 (deg pre-filled with 1.0 selfloop) --
__global__ void k_degree(const int* __restrict__ ei, int E, float* __restrict__ deg) {
    int e = blockIdx.x * blockDim.x + threadIdx.x;
    if (e < E) {
        int d = ei[E + e];
        unsafeAtomicAdd(&deg[d], 1.0f);
    }
}

// ---------------- dinv = rsqrt(deg), in place -------------------------------
__global__ void k_rsqrt(float* __restrict__ deg, int n) {
    int i = blockIdx.x * blockDim.x + threadIdx.x;
    if (i < n) deg[i] = rsqrtf(deg[i]);
}

// ---------------- layer-1 GEMM: y1 = (dinv*x) @ W1, 12 -> 16 ----------------
// One wave per 16-node tile.  K=12 -> three V_WMMA_F32_16X16X4_F32.
// A layout (16x4 f32): lanes 0-15 row M=lane,    VGPR0=K0 VGPR1=K1;
//                      lanes 16-31 row M=lane-16, VGPR0=K2 VGPR1=K3.
// B layout (4x16 f32): VGPR0 = rows K0/K2, VGPR1 = rows K1/K3, col = lane&15.
// C/D (16x16 f32): VGPR j -> row j (lanes 0-15) / row j+8 (lanes 16-31).
// dinv row-scaling is applied to A (one dinv load per lane).
__global__ void k_xw1_wmma(const float* __restrict__ x, const float* __restrict__ W,
                           const float* __restrict__ dinv, float* __restrict__ y,
                           int ntiles, int nnodes) {
    int lane = threadIdx.x & 31;
    int wave = threadIdx.x >> 5;
    int tile = blockIdx.x * (blockDim.x >> 5) + wave;
    if (tile >= ntiles) return;              // wave-uniform: EXEC stays all-1s
    int half = lane >> 4;                    // 0 or 1
    int l15  = lane & 15;
    int row  = tile * 16 + l15;
    bool full = (tile * 16 + 16) <= nnodes;  // wave-uniform
    int rowc = full ? row : (row < nnodes ? row : nnodes - 1);
    float di = dinv[rowc];

    v8f c = {};
#pragma unroll
    for (int s = 0; s < 3; ++s) {
        int kb = s * 4 + half * 2;
        v2f a, b;
        a.x = di * x[rowc * 12 + kb + 0];
        a.y = di * x[rowc * 12 + kb + 1];
        b.x = W[(kb + 0) * 16 + l15];
        b.y = W[(kb + 1) * 16 + l15];
        c = __builtin_amdgcn_wmma_f32_16x16x4_f32(false, a, false, b,
                                                  (short)0, c, false, false);
    }
    if (full) {
#pragma unroll
        for (int j = 0; j < 8; ++j)
            y[(size_t)(tile * 16 + j + half * 8) * 16 + l15] = c[j];
    } else {
#pragma unroll
        for (int j = 0; j < 8; ++j) {
            int node = tile * 16 + j + half * 8;
            if (node < nnodes) y[(size_t)node * 16 + l15] = c[j];
        }
    }
}

// ---------------- layer-2 GEMM: y2 = (dinv*h1) @ W2, 16 -> 8 (pad to 16) ----
__global__ void k_xw2_wmma(const float* __restrict__ h, const float* __restrict__ W2,
                           const float* __restrict__ dinv, float* __restrict__ y,
                           int ntiles, int nnodes) {
    int lane = threadIdx.x & 31;
    int wave = threadIdx.x >> 5;
    int tile = blockIdx.x * (blockDim.x >> 5) + wave;
    if (tile >= ntiles) return;
    int half = lane >> 4;
    int l15  = lane & 15;
    int row  = tile * 16 + l15;
    bool full = (tile * 16 + 16) <= nnodes;
    int rowc = full ? row : (row < nnodes ? row : nnodes - 1);
    float di = dinv[rowc];

    v8f c = {};
#pragma unroll
    for (int s = 0; s < 4; ++s) {
        int kb = s * 4 + half * 2;
        v2f a, b;
        a.x = di * h[(size_t)rowc * 16 + kb + 0];
        a.y = di * h[(size_t)rowc * 16 + kb + 1];
        b.x = (l15 < 8) ? W2[(kb + 0) * 8 + l15] : 0.0f;   // pad cols 8..15
        b.y = (l15 < 8) ? W2[(kb + 1) * 8 + l15] : 0.0f;
        c = __builtin_amdgcn_wmma_f32_16x16x4_f32(false, a, false, b,
                                                  (short)0, c, false, false);
    }
    if (full) {
        if (l15 < 8) {
#pragma unroll
            for (int j = 0; j < 8; ++j)
                y[(size_t)(tile * 16 + j + half * 8) * 8 + l15] = c[j];
        }
    } else {
        if (l15 < 8) {
#pragma unroll
            for (int j = 0; j < 8; ++j) {
                int node = tile * 16 + j + half * 8;
                if (node < nnodes) y[(size_t)node * 8 + l15] = c[j];
            }
        }
    }
}

// ---------------- edge scatter, 16 channels: agg[dst] += y[src] -------------
// One thread per (edge, 4-float chunk): float4 gather + 4 atomic f32 adds.
__global__ void k_edge16(const int* __restrict__ ei, int E,
                         const float* __restrict__ y, float* __restrict__ agg) {
    int t = blockIdx.x * blockDim.x + threadIdx.x;
    int e = t >> 2, part = t & 3;
    if (e >= E) return;
    int s = ei[e], d = ei[E + e];
    float4 v = *(const float4*)(y + (size_t)s * 16 + part * 4);
    float* a = agg + (size_t)d * 16 + part * 4;
    unsafeAtomicAdd(a + 0, v.x);
    unsafeAtomicAdd(a + 1, v.y);
    unsafeAtomicAdd(a + 2, v.z);
    unsafeAtomicAdd(a + 3, v.w);
}

// ---------------- edge scatter, 8 channels ----------------------------------
__global__ void k_edge8(const int* __restrict__ ei, int E,
                        const float* __restrict__ y, float* __restrict__ agg) {
    int t = blockIdx.x * blockDim.x + threadIdx.x;
    int e = t >> 1, part = t & 1;
    if (e >= E) return;
    int s = ei[e], d = ei[E + e];
    float4 v = *(const float4*)(y + (size_t)s * 8 + part * 4);
    float* a = agg + (size_t)d * 8 + part * 4;
    unsafeAtomicAdd(a + 0, v.x);
    unsafeAtomicAdd(a + 1, v.y);
    unsafeAtomicAdd(a + 2, v.z);
    unsafeAtomicAdd(a + 3, v.w);
}

// ---------------- layer-1 epilogue: h1 = relu(dinv*(agg+y) + b1) (in place) -
__global__ void k_post1(const float* __restrict__ y, float* __restrict__ agg,
                        const float* __restrict__ dinv, const float* __restrict__ b,
                        int n) {
    int i = blockIdx.x * blockDim.x + threadIdx.x;
    if (i >= n) return;
    float di = dinv[i];
#pragma unroll
    for (int c = 0; c < 16; c += 4) {
        float4 av = *(const float4*)(agg + (size_t)i * 16 + c);
        float4 yv = *(const float4*)(y + (size_t)i * 16 + c);
        float4 r;
        r.x = fmaxf(di * (av.x + yv.x) + b[c + 0], 0.0f);
        r.y = fmaxf(di * (av.y + yv.y) + b[c + 1], 0.0f);
        r.z = fmaxf(di * (av.z + yv.z) + b[c + 2], 0.0f);
        r.w = fmaxf(di * (av.w + yv.w) + b[c + 3], 0.0f);
        *(float4*)(agg + (size_t)i * 16 + c) = r;
    }
}

// ---------------- layer-2 epilogue + FC + sigmoid ---------------------------
__global__ void k_post2(const float* __restrict__ y, const float* __restrict__ agg,
                        const float* __restrict__ dinv, const float* __restrict__ b,
                        const float* __restrict__ fcw, const float* __restrict__ fcb,
                        float* __restrict__ out, int n) {
    int i = blockIdx.x * blockDim.x + threadIdx.x;
    if (i >= n) return;
    float di = dinv[i];
    float acc = fcb[0];
#pragma unroll
    for (int c = 0; c < 8; ++c) {
        float v = fmaxf(di * (agg[(size_t)i * 8 + c] + y[(size_t)i * 8 + c]) + b[c], 0.0f);
        acc += v * fcw[c];
    }
    out[i] = 1.0f / (1.0f + __expf(-acc));
}

// ---------------------------------------------------------------------------
extern "C" void kernel_launch(void* const* d_in, const int* in_sizes, int n_in,
                              void* d_out, int out_size, void* d_ws, size_t ws_size,
                              hipStream_t stream) {
    const float* x   = (const float*)d_in[0];
    const int*   ei  = (const int*)d_in[1];   // [2, E] int32
    const float* W1  = (const float*)d_in[2];
    const float* b1  = (const float*)d_in[3];
    const float* W2  = (const float*)d_in[4];
    const float* b2  = (const float*)d_in[5];
    const float* fcw = (const float*)d_in[6];
    const float* fcb = (const float*)d_in[7];
    float*       out = (float*)d_out;

    const int N = in_sizes[0] / 12;
    const int E = in_sizes[1] / 2;

    // workspace layout (floats), 33*N total (~66 MB @ N=500K):
    float* ws   = (float*)d_ws;
    float* dinv = ws;                       // [N]   deg -> dinv in place
    float* y1   = ws + (size_t)N;           // [16N] (y2 reuses first 8N later)
    float* agg1 = ws + (size_t)17 * N;      // [16N] agg1 -> h1 in place; agg2 reuses first 8N
    float* y2   = y1;
    float* agg2 = agg1;

    const int ntiles = (N + 15) / 16;
    const int gN = (N + TPB - 1) / TPB;
    const int gE = (E + TPB - 1) / TPB;
    const int gT = (ntiles + (TPB / 32) - 1) / (TPB / 32);

    // 1) deg = 1 (self loop), agg1 = 0
    k_fill_f32<<<gN, TPB, 0, stream>>>(dinv, 1.0f, N);
    k_fill_f32<<<(16 * N + TPB - 1) / TPB, TPB, 0, stream>>>(agg1, 0.0f, 16 * N);
    // 2) degree + dinv
    k_degree<<<gE, TPB, 0, stream>>>(ei, E, dinv);
    k_rsqrt<<<gN, TPB, 0, stream>>>(dinv, N);
    // 3) layer 1: y1 = (dinv*x)@W1; scatter; epilogue (h1 in agg1)
    k_xw1_wmma<<<gT, TPB, 0, stream>>>(x, W1, dinv, y1, ntiles, N);
    k_edge16<<<(4 * E + TPB - 1) / TPB, TPB, 0, stream>>>(ei, E, y1, agg1);
    k_post1<<<gN, TPB, 0, stream>>>(y1, agg1, dinv, b1, N);
    // 4) layer 2: y2 = (dinv*h1)@W2  (reads agg1-as-h1, writes over y1 region)
    k_xw2_wmma<<<gT, TPB, 0, stream>>>(agg1, W2, dinv, y2, ntiles, N);
    // agg2 reuses agg1's first 8N only after h1 has been consumed
    k_fill_f32<<<(8 * N + TPB - 1) / TPB, TPB, 0, stream>>>(agg2, 0.0f, 8 * N);
    k_edge8<<<(2 * E + TPB - 1) / TPB, TPB, 0, stream>>>(ei, E, y2, agg2);
    // 5) epilogue + fc + sigmoid
    k_post2<<<gN, TPB, 0, stream>>>(y2, agg2, dinv, b2, fcw, fcb, out, N);
}